// CoActivationComputer_86732569575517
// MI455X (gfx1250) — compile-verified
//
#include <hip/hip_runtime.h>
#include <hip/hip_bf16.h>

typedef __attribute__((ext_vector_type(2))) float v2f;
typedef __attribute__((ext_vector_type(4))) float v4f;
typedef __attribute__((ext_vector_type(8))) float v8f;

#define B_  32
#define N_  2048
#define C_  64
#define R_  32
#define K_  64
#define RC_ (R_ * C_)   // 2048, == N_

// ---------------------------------------------------------------------------
// Kernel 1: per-row top-64 (value-desc, smallest-index tie-break like
// jax.lax.top_k) + masked dense row W (top-k entries keep value, rest 0).
// One block per batch row; row lives in LDS.
// ---------------------------------------------------------------------------
__global__ __launch_bounds__(256) void topk_kernel(const float* __restrict__ acc,
                                                   float* __restrict__ out_vals,
                                                   float* __restrict__ out_idx,
                                                   float* __restrict__ W)
{
    __shared__ float sv[N_];
    __shared__ float rv[256];
    __shared__ int   ri[256];

    const int b   = blockIdx.x;
    const int tid = threadIdx.x;
    const float* row = acc + b * N_;

    for (int i = tid; i < N_; i += 256) sv[i] = row[i];
    __syncthreads();

    for (int t = 0; t < K_; ++t) {
        // local argmax over strided slice; strict '>' keeps smallest index
        float bv = -__builtin_inff();
        int   bi = 0x7fffffff;
        for (int i = tid; i < N_; i += 256) {
            float v = sv[i];
            if (v > bv) { bv = v; bi = i; }
        }
        rv[tid] = bv; ri[tid] = bi;
        __syncthreads();
        for (int s = 128; s > 0; s >>= 1) {
            if (tid < s) {
                float v2 = rv[tid + s]; int i2 = ri[tid + s];
                if (v2 > rv[tid] || (v2 == rv[tid] && i2 < ri[tid])) {
                    rv[tid] = v2; ri[tid] = i2;
                }
            }
            __syncthreads();
        }
        if (tid == 0) {
            out_vals[b * K_ + t] = rv[0];
            out_idx [b * K_ + t] = (float)ri[0];
            sv[ri[0]] = -__builtin_inff();     // mark selected
        }
        __syncthreads();
    }

    // masked row: selected entries were overwritten with -inf in sv
    for (int i = tid; i < N_; i += 256) {
        W[b * N_ + i] = (sv[i] == -__builtin_inff()) ? row[i] : 0.0f;
    }
}

// ---------------------------------------------------------------------------
// Kernel 2: pack H[n, r*64 + c] = U[c, n, r]   (U is [C, N, R])
// Output-coalesced transpose; 16 MB, stays resident in the 192 MB L2.
// ---------------------------------------------------------------------------
__global__ __launch_bounds__(256) void pack_kernel(const float* __restrict__ U,
                                                   float* __restrict__ H)
{
    int o = blockIdx.x * 256 + threadIdx.x;          // [0, N_*RC_)
    if (o >= N_ * RC_) return;
    int c = o & (C_ - 1);
    int r = (o >> 6) & (R_ - 1);
    int n = o >> 11;
    H[o] = U[(size_t)c * (N_ * R_) + (size_t)n * R_ + r];
}

// ---------------------------------------------------------------------------
// Kernel 3: Au = W @ H   (M=32, N=2048, K=2048) with V_WMMA_F32_16X16X4_F32.
// One wave per 16x16 output tile (2 x 128 tiles = 256 waves).
// ISA 7.12.2 f32 layouts:
//   A 16x4 : lanes 0-15 -> M=lane, v0=K0,v1=K1 ; lanes 16-31 -> v0=K2,v1=K3
//   B 4x16 : v0 = rows K0(lo half)/K2(hi half), v1 = K1/K3, N = lane&15
//   C/D    : acc[v] -> row m0 + v + 8*(lane>=16), col n0 + (lane&15)
// ---------------------------------------------------------------------------
__global__ __launch_bounds__(256) void gemm_wmma_kernel(const float* __restrict__ W,
                                                        const float* __restrict__ H,
                                                        float* __restrict__ Au)
{
    const int wave = (blockIdx.x * 256 + (int)threadIdx.x) >> 5;   // 0..255
    const int lane = threadIdx.x & 31;
    const int mt   = wave >> 7;          // 0..1
    const int nt   = wave & 127;         // 0..127
    const int m0   = mt * 16;
    const int n0   = nt * 16;
    const int half = lane >> 4;          // 0 or 1
    const int ml   = lane & 15;

    const float* wrow = W + (m0 + ml) * N_;      // A operand row
    const float* hcol = H + n0 + ml;             // B operand column

    v8f acc = {};
    for (int k = 0; k < N_; k += 4) {
        const int k0 = k + half * 2;
        v2f a, bm;
        a.x  = wrow[k0];
        a.y  = wrow[k0 + 1];
        bm.x = hcol[(size_t)k0 * RC_];
        bm.y = hcol[(size_t)(k0 + 1) * RC_];
        acc = __builtin_amdgcn_wmma_f32_16x16x4_f32(
                  /*neg_a=*/false, a, /*neg_b=*/false, bm,
                  /*c_mod=*/(short)0, acc, /*reuse_a=*/false, /*reuse_b=*/false);
    }

    const int col   = n0 + ml;
    const int rbase = m0 + half * 8;
#pragma unroll
    for (int v = 0; v < 8; ++v) {
        Au[(rbase + v) * RC_ + col] = acc[v];
    }
}

// ---------------------------------------------------------------------------
// Kernel 4: frob[b,c] = sum_r Au[b, r*64+c]^2 / (K*R + 1e-9)
// ---------------------------------------------------------------------------
__global__ __launch_bounds__(256) void frob_kernel(const float* __restrict__ Au,
                                                   float* __restrict__ frob)
{
    int o = blockIdx.x * 256 + threadIdx.x;      // [0, B_*C_)
    if (o >= B_ * C_) return;
    int b = o >> 6;
    int c = o & (C_ - 1);
    const float* p = Au + b * RC_ + c;
    float s = 0.0f;
#pragma unroll
    for (int r = 0; r < R_; ++r) {
        float v = p[r * C_];
        s += v * v;
    }
    frob[o] = s / ((float)(K_ * R_) + 1e-9f);
}

// ---------------------------------------------------------------------------
// Kernel 5: M[b,i,j] = W[b,i] * W[b,j]  (distinct top-k indices => scatter-add
// == masked outer product). 512 MB streamed out with non-temporal stores.
// ---------------------------------------------------------------------------
__global__ __launch_bounds__(256) void outer_kernel(const float* __restrict__ W,
                                                    float* __restrict__ M)
{
    const size_t total  = (size_t)B_ * N_ * (N_ / 4);   // float4 count
    const size_t stride = (size_t)gridDim.x * 256;
    for (size_t t = (size_t)blockIdx.x * 256 + threadIdx.x; t < total; t += stride) {
        size_t b   = t / ((size_t)N_ * (N_ / 4));
        size_t rem = t % ((size_t)N_ * (N_ / 4));
        size_t i   = rem / (N_ / 4);
        size_t j4  = rem % (N_ / 4);

        const float wi = W[b * N_ + i];
        const v4f  wj = *((const v4f*)(W + b * N_) + j4);
        v4f o = wi * wj;
        __builtin_nontemporal_store(o, (v4f*)M + t);
    }
}

// ---------------------------------------------------------------------------
extern "C" void kernel_launch(void* const* d_in, const int* in_sizes, int n_in,
                              void* d_out, int out_size, void* d_ws, size_t ws_size,
                              hipStream_t stream)
{
    const float* acc = (const float*)d_in[0];   // [B, N]
    const float* U   = (const float*)d_in[1];   // [C, N, R]

    float* out      = (float*)d_out;
    float* out_vals = out;                       // [B, K]
    float* out_idx  = out + B_ * K_;             // [B, K] (indices as float)
    float* out_frob = out + 2 * B_ * K_;         // [B, C]
    float* out_M    = out + 3 * B_ * K_;         // [B, N, N]

    char* ws = (char*)d_ws;
    float* W  = (float*)ws;                                              // 256 KB
    float* H  = (float*)(ws + (size_t)B_ * N_ * sizeof(float));          // 16 MB
    float* Au = (float*)(ws + (size_t)B_ * N_ * sizeof(float)
                            + (size_t)N_ * RC_ * sizeof(float));         // 256 KB

    topk_kernel<<<B_, 256, 0, stream>>>(acc, out_vals, out_idx, W);

    pack_kernel<<<(N_ * RC_) / 256, 256, 0, stream>>>(U, H);

    gemm_wmma_kernel<<<32, 256, 0, stream>>>(W, H, Au);   // 256 waves, 256 tiles

    frob_kernel<<<(B_ * C_ + 255) / 256, 256, 0, stream>>>(Au, out_frob);

    outer_kernel<<<16384, 256, 0, stream>>>(W, out_M);
}